// ConcatenationUnit_3607772528791
// MI455X (gfx1250) — compile-verified
//
#include <hip/hip_runtime.h>
#include <math.h>

typedef float v2f __attribute__((ext_vector_type(2)));
typedef float v4f __attribute__((ext_vector_type(4)));
typedef float v8f __attribute__((ext_vector_type(8)));

#define NROWS 4096
#define CIN   1024
#define INTER 128
#define RPIX  16384   // 128*128
#define GCH   256
#define KB    64      // K-block staged in LDS for the pairwise matmul
#define LPAD  8       // LDS row padding: row stride 136 -> lane halves hit disjoint banks

// ---------------------------------------------------------------------------
// u[k] = sum_c cw[cw_off+c] * W[c, k]   (W: [128, K] row-major)
// cout  = sum_c cw[cw_off+c] * bias[c]
// ---------------------------------------------------------------------------
__global__ void fold_u_kernel(const float* __restrict__ W, const float* __restrict__ bias,
                              const float* __restrict__ cw, int cw_off, int K,
                              float* __restrict__ u, float* __restrict__ cout) {
  int k = blockIdx.x * blockDim.x + threadIdx.x;
  if (k < K) {
    float s = 0.f;
    for (int c = 0; c < INTER; ++c) s += cw[cw_off + c] * W[(size_t)c * K + k];
    u[k] = s;
  }
  if (blockIdx.x == 0 && threadIdx.x == 0) {
    float s = 0.f;
    for (int c = 0; c < INTER; ++c) s += cw[cw_off + c] * bias[c];
    *cout = s;
  }
}

// ---------------------------------------------------------------------------
// xbar[k] = mean over rows of concat(A, B)  (each [rows, K])
// ---------------------------------------------------------------------------
__global__ void col_mean2_kernel(const float* __restrict__ A, const float* __restrict__ B,
                                 int rows, int K, float* __restrict__ xbar) {
  int k = blockIdx.x * blockDim.x + threadIdx.x;
  if (k >= K) return;
  float s = 0.f;
  for (int i = 0; i < rows; ++i)
    s += A[(size_t)i * K + k] + B[(size_t)i * K + k];
  xbar[k] = s / (2.0f * (float)rows);
}

// ---------------------------------------------------------------------------
// wf[gc*9+d] = sum_c cw[128+c] * sp_pw[c, gc*9+d];  cf = sum_c cw[128+c]*sp_pb[c]
// ---------------------------------------------------------------------------
__global__ void fold_conv_kernel(const float* __restrict__ sp_pw, const float* __restrict__ sp_pb,
                                 const float* __restrict__ cw, float* __restrict__ wf,
                                 float* __restrict__ cf) {
  int idx = blockIdx.x * blockDim.x + threadIdx.x;
  if (idx < GCH * 9) {
    float s = 0.f;
    for (int c = 0; c < INTER; ++c) s += cw[INTER + c] * sp_pw[(size_t)c * (GCH * 9) + idx];
    wf[idx] = s;
  }
  if (blockIdx.x == 0 && threadIdx.x == 0) {
    float s = 0.f;
    for (int c = 0; c < INTER; ++c) s += cw[INTER + c] * sp_pb[c];
    *cf = s;
  }
}

// ---------------------------------------------------------------------------
// Per-row dot(s):  out[r] = maybe_relu( X[r]*u1 + *c1 + (Y ? Y[r]*u2 + *c2 : 0) + (ex?*ex:0) )
// one wave per row, 8 waves / 256-thread block
// ---------------------------------------------------------------------------
__global__ void row_dot2_kernel(const float* __restrict__ X, const float* __restrict__ u1,
                                const float* __restrict__ c1,
                                const float* __restrict__ Y, const float* __restrict__ u2,
                                const float* __restrict__ c2,
                                const float* __restrict__ ex,
                                int do_relu, int K, int rows, float* __restrict__ out) {
  int wave = threadIdx.x >> 5, lane = threadIdx.x & 31;
  int row = blockIdx.x * 8 + wave;
  if (row >= rows) return;
  const float* xr = X + (size_t)row * K;
  float s = 0.f;
  for (int k = lane; k < K; k += 32) s += xr[k] * u1[k];
  if (Y) {
    const float* yr = Y + (size_t)row * K;
    for (int k = lane; k < K; k += 32) s += yr[k] * u2[k];
  }
  for (int off = 16; off > 0; off >>= 1) s += __shfl_down(s, off, 32);
  if (lane == 0) {
    s += c1[0];
    if (c2) s += c2[0];
    if (ex) s += ex[0];
    out[row] = do_relu ? fmaxf(s, 0.f) : s;
  }
}

// ---------------------------------------------------------------------------
// out[r] = W[r,:]*v + bias[r]   (W: [rows, K]); one wave per row
// ---------------------------------------------------------------------------
__global__ void row_matvec_kernel(const float* __restrict__ W, const float* __restrict__ v,
                                  const float* __restrict__ bias, float* __restrict__ out,
                                  int K, int rows) {
  int wave = threadIdx.x >> 5, lane = threadIdx.x & 31;
  int row = blockIdx.x * 8 + wave;
  if (row >= rows) return;
  const float* wr = W + (size_t)row * K;
  float s = 0.f;
  for (int k = lane; k < K; k += 32) s += wr[k] * v[k];
  for (int off = 16; off > 0; off >>= 1) s += __shfl_down(s, off, 32);
  if (lane == 0) out[row] = s + bias[row];
}

// single short dot: out = x . y[yoff..]
__global__ void dot_small_kernel(const float* __restrict__ x, const float* __restrict__ y,
                                 int yoff, int K, float* __restrict__ out) {
  int lane = threadIdx.x;
  float s = 0.f;
  for (int k = lane; k < K; k += 32) s += x[k] * y[yoff + k];
  for (int off = 16; off > 0; off >>= 1) s += __shfl_down(s, off, 32);
  if (lane == 0) out[0] = s;
}

// ---------------------------------------------------------------------------
// single-channel 3x3 SAME conv:  bs[h,w] = cf + sum_{gc,ky,kx} wf[gc,ky,kx]*G[gc,h+ky-1,w+kx-1]
// ---------------------------------------------------------------------------
__global__ void conv1_kernel(const float* __restrict__ G, const float* __restrict__ wf,
                             const float* __restrict__ cf, float* __restrict__ bs) {
  int p = blockIdx.x * blockDim.x + threadIdx.x;
  if (p >= RPIX) return;
  int h = p >> 7, w = p & 127;
  float s = cf[0];
  for (int gc = 0; gc < GCH; ++gc) {
    const float* g = G + (size_t)gc * RPIX;
    const float* wv = wf + gc * 9;
#pragma unroll
    for (int ky = 0; ky < 3; ++ky) {
      int hh = h + ky - 1;
      if (hh < 0 || hh >= 128) continue;
#pragma unroll
      for (int kx = 0; kx < 3; ++kx) {
        int ww = w + kx - 1;
        if (ww < 0 || ww >= 128) continue;
        s += wv[ky * 3 + kx] * g[hh * 128 + ww];
      }
    }
  }
  bs[p] = s;
}

// ---------------------------------------------------------------------------
// softmax over 16384 values, single workgroup of 1024 threads
// ---------------------------------------------------------------------------
__global__ void softmax16k_kernel(const float* __restrict__ bs, float* __restrict__ sw) {
  __shared__ float red[1024];
  int t = threadIdx.x;
  float m = -3.0e38f;
  for (int j = t; j < RPIX; j += 1024) m = fmaxf(m, bs[j]);
  red[t] = m; __syncthreads();
  for (int s = 512; s > 0; s >>= 1) { if (t < s) red[t] = fmaxf(red[t], red[t + s]); __syncthreads(); }
  m = red[0]; __syncthreads();
  float sum = 0.f;
  for (int j = t; j < RPIX; j += 1024) sum += __expf(bs[j] - m);
  red[t] = sum; __syncthreads();
  for (int s = 512; s > 0; s >>= 1) { if (t < s) red[t] += red[t + s]; __syncthreads(); }
  float inv = 1.0f / red[0];
  for (int j = t; j < RPIX; j += 1024) sw[j] = __expf(bs[j] - m) * inv;
}

// ---------------------------------------------------------------------------
// t[gc*9 + ky*3+kx] = sum_{h,w} sw[h,w] * G[gc, h+ky-1, w+kx-1]  (OOB -> 0)
// one wave per output
// ---------------------------------------------------------------------------
__global__ void tred_kernel(const float* __restrict__ G, const float* __restrict__ sw,
                            float* __restrict__ t) {
  int wave = threadIdx.x >> 5, lane = threadIdx.x & 31;
  int o = blockIdx.x * 8 + wave;
  if (o >= GCH * 9) return;
  int gc = o / 9, d = o % 9;
  int dy = d / 3 - 1, dx = d % 3 - 1;
  const float* g = G + (size_t)gc * RPIX;
  float s = 0.f;
  for (int p = lane; p < RPIX; p += 32) {
    int h = (p >> 7) + dy, w = (p & 127) + dx;
    if (h >= 0 && h < 128 && w >= 0 && w < 128) s += sw[p] * g[h * 128 + w];
  }
  for (int off = 16; off > 0; off >>= 1) s += __shfl_down(s, off, 32);
  if (lane == 0) t[o] = s;
}

// ---------------------------------------------------------------------------
// WMMA f32 GEMM:  Out[M,128] = X[M,K] @ W[128,K]^T + bias
// block = 256 thr = 8 waves; block covers 16 rows, each wave one 16-col tile.
// b128 loads: per K-step of 8, lane loads float4 at base k + 4*(lane>>4);
// wmma#1 eats (.x,.y) -> K {k,k+1,k+4,k+5}, wmma#2 eats (.z,.w) -> {k+2,k+3,k+6,k+7}.
// Any lane->K permutation is valid as long as A and B use the same one.
// ---------------------------------------------------------------------------
__global__ void wmma_gemm_xwT_kernel(const float* __restrict__ X, const float* __restrict__ W,
                                     const float* __restrict__ bias, float* __restrict__ Out,
                                     int K) {
  int wave = threadIdx.x >> 5, lane = threadIdx.x & 31;
  int m0 = blockIdx.x * 16;
  int n0 = wave * 16;
  int mi = lane & 15;
  int h4 = (lane >> 4) << 2;  // per-lane float4 base offset: 0 or 4
  const float* xrow = X + (size_t)(m0 + mi) * K + h4;
  const float* wrow = W + (size_t)(n0 + mi) * K + h4;
  float bvv = bias[n0 + mi];
  v8f acc;
#pragma unroll
  for (int v = 0; v < 8; ++v) acc[v] = bvv;
#pragma unroll 4
  for (int k = 0; k < K; k += 8) {
    v4f a4 = *(const v4f*)(xrow + k);
    v4f b4 = *(const v4f*)(wrow + k);
    v2f A1; A1.x = a4.x; A1.y = a4.y;
    v2f B1; B1.x = b4.x; B1.y = b4.y;
    acc = __builtin_amdgcn_wmma_f32_16x16x4_f32(false, A1, false, B1, (short)0, acc,
                                                false, false);
    v2f A2; A2.x = a4.z; A2.y = a4.w;
    v2f B2; B2.x = b4.z; B2.y = b4.w;
    acc = __builtin_amdgcn_wmma_f32_16x16x4_f32(false, A2, false, B2, (short)0, acc,
                                                false, false);
  }
  int hi = lane >> 4;
  float* orow = Out + (size_t)(m0 + 8 * hi) * INTER + n0 + mi;
#pragma unroll
  for (int v = 0; v < 8; ++v) orow[(size_t)v * INTER] = acc[v];
}

// ---------------------------------------------------------------------------
// origin_rel[i,n] = sum_j relu(a[i]+b[j]) * (1/N) * gx[j,n]
// A generated in registers (rank-structured pairwise matrix).
// B (gx) staged in LDS per 64-row K-block: the 8 waves of a block share the
// same K-rows, so they stage them cooperatively (2MB of gx per block, 512MB total).
// LDS row stride 136 floats: lane-half row offset 2 rows = 272 banks%64 = 16,
// so halves read disjoint bank sets (no conflicts).
// ---------------------------------------------------------------------------
__global__ void wmma_origin_kernel(const float* __restrict__ a, const float* __restrict__ b,
                                   const float* __restrict__ gx, float* __restrict__ Out) {
  __shared__ float lgx[KB][INTER + LPAD];
  int wave = threadIdx.x >> 5, lane = threadIdx.x & 31;
  int m0 = blockIdx.x * 16;
  int n0 = wave * 16;
  int mi = lane & 15;
  int kk = (lane >> 4) << 1;
  const float inv = 1.0f / (float)NROWS;
  float areg = a[m0 + mi];
  v8f acc;
#pragma unroll
  for (int v = 0; v < 8; ++v) acc[v] = 0.f;
  for (int kb = 0; kb < NROWS; kb += KB) {
    // cooperative stage of KB x 128 floats: 8 float4 per thread
    for (int i = threadIdx.x * 4; i < KB * INTER; i += 256 * 4) {
      int r = i >> 7, c = i & 127;
      *(v4f*)&lgx[r][c] = *(const v4f*)&gx[(size_t)(kb + r) * INTER + c];
    }
    __syncthreads();
#pragma unroll 4
    for (int j = 0; j < KB; j += 4) {
      float b0 = b[kb + j + kk], b1 = b[kb + j + kk + 1];
      v2f A;
      A.x = fmaxf(areg + b0, 0.f) * inv;
      A.y = fmaxf(areg + b1, 0.f) * inv;
      v2f B;
      B.x = lgx[j + kk][n0 + mi];
      B.y = lgx[j + kk + 1][n0 + mi];
      acc = __builtin_amdgcn_wmma_f32_16x16x4_f32(false, A, false, B, (short)0, acc,
                                                  false, false);
    }
    __syncthreads();
  }
  int hi = lane >> 4;
  float* orow = Out + (size_t)(m0 + 8 * hi) * INTER + n0 + mi;
#pragma unroll
  for (int v = 0; v < 8; ++v) orow[(size_t)v * INTER] = acc[v];
}

// ---------------------------------------------------------------------------
// out += 3 pair terms + broadcast spatial vector + before/after rank-1 term
// ---------------------------------------------------------------------------
__global__ void combine_kernel(float* __restrict__ out,
                               const float* __restrict__ gy0, const float* __restrict__ gy1,
                               const float* __restrict__ gy2,
                               const float* __restrict__ s0, const float* __restrict__ s1,
                               const float* __restrict__ s2,
                               const float* __restrict__ pr_g,
                               const float* __restrict__ vsp, const float* __restrict__ spg,
                               const float* __restrict__ sba, const float* __restrict__ gm,
                               const float* __restrict__ bag) {
  int idx = blockIdx.x * blockDim.x + threadIdx.x;
  if (idx >= NROWS * INTER) return;
  int i = idx >> 7, n = idx & 127;
  float r = out[idx];
  r += pr_g[0] * s0[i] * gy0[idx];
  r += pr_g[1] * s1[i] * gy1[idx];
  r += pr_g[2] * s2[i] * gy2[idx];
  r += spg[0] * vsp[n];
  r += bag[0] * sba[i] * gm[n];
  out[idx] = r;
}

// ---------------------------------------------------------------------------
extern "C" void kernel_launch(void* const* d_in, const int* in_sizes, int n_in,
                              void* d_out, int out_size, void* d_ws, size_t ws_size,
                              hipStream_t stream) {
  const float* x    = (const float*)d_in[0];
  const float* loc  = (const float*)d_in[1];
  const float* G    = (const float*)d_in[2];
  const float* bef  = (const float*)d_in[3];
  const float* aft  = (const float*)d_in[4];
  const float* befl = (const float*)d_in[5];
  const float* aftl = (const float*)d_in[6];
  const float* m_gw = (const float*)d_in[9];
  const float* m_gb = (const float*)d_in[10];
  const float* m_tw = (const float*)d_in[11];
  const float* m_tb = (const float*)d_in[12];
  const float* m_pw = (const float*)d_in[13];
  const float* m_pb = (const float*)d_in[14];
  const float* m_cw = (const float*)d_in[15];
  const float* pr_gw = (const float*)d_in[16];
  const float* pr_gb = (const float*)d_in[17];
  const float* pr_tw = (const float*)d_in[18];
  const float* pr_tb = (const float*)d_in[19];
  const float* pr_pw = (const float*)d_in[20];
  const float* pr_pb = (const float*)d_in[21];
  const float* pr_cw = (const float*)d_in[22];
  const float* pr_g  = (const float*)d_in[23];
  const float* sp_gw = (const float*)d_in[24];
  const float* sp_gb = (const float*)d_in[25];
  const float* sp_pw = (const float*)d_in[26];
  const float* sp_pb = (const float*)d_in[27];
  const float* sp_cw = (const float*)d_in[30];
  const float* sp_g  = (const float*)d_in[31];
  const float* ba_gw = (const float*)d_in[32];
  const float* ba_gb = (const float*)d_in[33];
  const float* ba_tw = (const float*)d_in[34];
  const float* ba_tb = (const float*)d_in[35];
  const float* ba_pw = (const float*)d_in[36];
  const float* ba_pb = (const float*)d_in[37];
  const float* ba_cw = (const float*)d_in[38];
  const float* ba_g  = (const float*)d_in[39];
  float* out = (float*)d_out;

  float* w = (float*)d_ws;
  size_t off = 0;
  auto alloc = [&](size_t n) { float* p = w + off; off += n; return p; };

  float* gx   = alloc((size_t)NROWS * INTER);
  float* gy0  = alloc((size_t)NROWS * INTER);
  float* gy1  = alloc((size_t)NROWS * INTER);
  float* gy2  = alloc((size_t)NROWS * INTER);
  float* av   = alloc(NROWS);
  float* bv   = alloc(NROWS);
  float* s0   = alloc(NROWS);
  float* s1   = alloc(NROWS);
  float* s2   = alloc(NROWS);
  float* sba  = alloc(NROWS);
  float* u_m  = alloc(CIN);
  float* u_p  = alloc(CIN);
  float* u_t0 = alloc(CIN);
  float* u_t1 = alloc(CIN);
  float* u_t2 = alloc(CIN);
  float* u_q0 = alloc(CIN);
  float* u_q1 = alloc(CIN);
  float* u_q2 = alloc(CIN);
  float* u_ba = alloc(CIN);
  float* xbar = alloc(CIN);
  float* gm   = alloc(INTER);
  float* pm   = alloc(INTER);
  float* wf   = alloc(GCH * 9);
  float* bs   = alloc(RPIX);
  float* sw   = alloc(RPIX);
  float* tt   = alloc(GCH * 9);
  float* vsp  = alloc(INTER);
  float* sc   = alloc(16);  // [0]=c_a [1]=c_b [2..4]=c_t [5..7]=c_p [8]=c_ba [9]=pmdot [10]=c_fold
  (void)ws_size; (void)in_sizes; (void)n_in; (void)out_size;

  dim3 b256(256);
  // --- folds (independent, cheap) ---
  fold_u_kernel<<<4, b256, 0, stream>>>(m_tw, m_tb, m_cw, 0, CIN, u_m, sc + 0);
  fold_u_kernel<<<4, b256, 0, stream>>>(m_pw, m_pb, m_cw, INTER, CIN, u_p, sc + 1);
  float* uts[3] = {u_t0, u_t1, u_t2};
  float* uqs[3] = {u_q0, u_q1, u_q2};
  for (int i = 0; i < 3; ++i) {
    fold_u_kernel<<<4, b256, 0, stream>>>(pr_tw + (size_t)i * INTER * CIN, pr_tb + i * INTER,
                                          pr_cw + i * 2 * INTER, 0, CIN, uts[i], sc + 2 + i);
    fold_u_kernel<<<4, b256, 0, stream>>>(pr_pw + (size_t)i * INTER * CIN, pr_pb + i * INTER,
                                          pr_cw + i * 2 * INTER, INTER, CIN, uqs[i], sc + 5 + i);
  }
  fold_u_kernel<<<4, b256, 0, stream>>>(ba_tw, ba_tb, ba_cw, 0, CIN, u_ba, sc + 8);
  col_mean2_kernel<<<4, b256, 0, stream>>>(bef, aft, NROWS, CIN, xbar);
  fold_conv_kernel<<<9, b256, 0, stream>>>(sp_pw, sp_pb, sp_cw, wf, sc + 10);

  // --- per-row scalars ---
  row_dot2_kernel<<<512, b256, 0, stream>>>(x, u_m, sc + 0, nullptr, nullptr, nullptr, nullptr,
                                            0, CIN, NROWS, av);
  row_dot2_kernel<<<512, b256, 0, stream>>>(x, u_p, sc + 1, nullptr, nullptr, nullptr, nullptr,
                                            0, CIN, NROWS, bv);
  const float* ys[3] = {loc, befl, aftl};
  float* ss[3] = {s0, s1, s2};
  for (int i = 0; i < 3; ++i)
    row_dot2_kernel<<<512, b256, 0, stream>>>(x, uts[i], sc + 2 + i, ys[i], uqs[i], sc + 5 + i,
                                              nullptr, 1, CIN, NROWS, ss[i]);
  // --- before/after (mean commutes with linear) ---
  row_matvec_kernel<<<16, b256, 0, stream>>>(ba_gw, xbar, ba_gb, gm, CIN, INTER);
  row_matvec_kernel<<<16, b256, 0, stream>>>(ba_pw, xbar, ba_pb, pm, CIN, INTER);
  dot_small_kernel<<<1, 32, 0, stream>>>(pm, ba_cw, INTER, INTER, sc + 9);
  row_dot2_kernel<<<512, b256, 0, stream>>>(x, u_ba, sc + 8, nullptr, nullptr, nullptr, sc + 9,
                                            1, CIN, NROWS, sba);
  // --- spatial (softmax rank-1 collapse) ---
  conv1_kernel<<<RPIX / 256, b256, 0, stream>>>(G, wf, sc + 10, bs);
  softmax16k_kernel<<<1, 1024, 0, stream>>>(bs, sw);
  tred_kernel<<<(GCH * 9 + 7) / 8, b256, 0, stream>>>(G, sw, tt);
  row_matvec_kernel<<<16, b256, 0, stream>>>(sp_gw, tt, sp_gb, vsp, GCH * 9, INTER);

  // --- WMMA GEMMs ---
  wmma_gemm_xwT_kernel<<<NROWS / 16, b256, 0, stream>>>(x, m_gw, m_gb, gx, CIN);
  wmma_gemm_xwT_kernel<<<NROWS / 16, b256, 0, stream>>>(loc, pr_gw, pr_gb, gy0, CIN);
  wmma_gemm_xwT_kernel<<<NROWS / 16, b256, 0, stream>>>(befl, pr_gw + (size_t)INTER * CIN,
                                                        pr_gb + INTER, gy1, CIN);
  wmma_gemm_xwT_kernel<<<NROWS / 16, b256, 0, stream>>>(aftl, pr_gw + (size_t)2 * INTER * CIN,
                                                        pr_gb + 2 * INTER, gy2, CIN);
  // --- big pairwise matmul, A in registers, B staged through LDS ---
  wmma_origin_kernel<<<NROWS / 16, b256, 0, stream>>>(av, bv, gx, out);
  // --- final combine ---
  combine_kernel<<<(NROWS * INTER) / 256, b256, 0, stream>>>(out, gy0, gy1, gy2, s0, s1, s2,
                                                             pr_g, vsp, sp_g, sba, gm, ba_g);
}